// GCNMV_64175401337157
// MI455X (gfx1250) — compile-verified
//
#include <hip/hip_runtime.h>
#include <math.h>

#define N_NODES 100000
#define N_EDGES 1600000
#define IN_F 48
#define HID_F 16
#define OUT_F 60

typedef __attribute__((ext_vector_type(2))) float v2f;
typedef __attribute__((ext_vector_type(8))) float v8f;

// ---------------------------------------------------------------------------
// GEMM (dual output): Y = X @ Wrel, R = X @ Wroot.  X:[N_NODES,K], W:[K,16].
// One wave computes one 16x16 tile (16 nodes x 16 out-features) via
// V_WMMA_F32_16X16X4_F32, looping K in steps of 4. Shared A-fragment feeds
// two accumulators (rel + root).
// ---------------------------------------------------------------------------
__global__ void gemm_dual16(const float* __restrict__ X, int K,
                            const float* __restrict__ Wrel,
                            const float* __restrict__ Wroot,
                            float* __restrict__ Y, float* __restrict__ R) {
    int gwave = (int)((blockIdx.x * blockDim.x + threadIdx.x) >> 5);
    if (gwave >= N_NODES / 16) return;        // wave-uniform: EXEC full for WMMA
    int lane = threadIdx.x & 31;
    int half = lane >> 4;                     // 0: K+0/1, 1: K+2/3
    int l    = lane & 15;                     // row (A) / col (B)
    int row0 = gwave * 16;
    const float* xrow = X + (size_t)(row0 + l) * K;

    v8f cy = {}; v8f cr = {};
    for (int k0 = 0; k0 < K; k0 += 4) {
        int kk = k0 + 2 * half;
        v2f a;  a.x  = xrow[kk];            a.y  = xrow[kk + 1];
        v2f br; br.x = Wrel[kk * 16 + l];   br.y = Wrel[(kk + 1) * 16 + l];
        v2f bo; bo.x = Wroot[kk * 16 + l];  bo.y = Wroot[(kk + 1) * 16 + l];
        cy = __builtin_amdgcn_wmma_f32_16x16x4_f32(false, a, false, br, (short)0, cy, false, false);
        cr = __builtin_amdgcn_wmma_f32_16x16x4_f32(false, a, false, bo, (short)0, cr, false, false);
    }
    int mb = half * 8;                        // D layout: VGPR i -> row mb+i, col l
    for (int i = 0; i < 8; ++i) {
        Y[(size_t)(row0 + mb + i) * 16 + l] = cy[i];
        R[(size_t)(row0 + mb + i) * 16 + l] = cr[i];
    }
}

// ---------------------------------------------------------------------------
// Layer-3 fused GEMM: OUT = AGG @ Wrel + H @ Wroot + bias.  K=16, N=60.
// N tiled by 16 (4 tiles); columns >= 60 masked (zero B-frag, no store).
// Bias pre-loaded into the accumulator (same value in all 8 row-VGPRs).
// ---------------------------------------------------------------------------
__global__ void gemm_out60(const float* __restrict__ AGG,
                           const float* __restrict__ H,
                           const float* __restrict__ Wrel,
                           const float* __restrict__ Wroot,
                           const float* __restrict__ bias,
                           float* __restrict__ OUT) {
    int gwave = (int)((blockIdx.x * blockDim.x + threadIdx.x) >> 5);
    int nt = gwave & 3;
    int mt = gwave >> 2;
    if (mt >= N_NODES / 16) return;           // wave-uniform
    int lane = threadIdx.x & 31;
    int half = lane >> 4;
    int l    = lane & 15;
    int row0 = mt * 16;
    int col  = nt * 16 + l;
    bool cv  = (col < OUT_F);

    float bv = cv ? bias[col] : 0.0f;
    v8f c; for (int i = 0; i < 8; ++i) c[i] = bv;

    const float* arow = AGG + (size_t)(row0 + l) * 16;
    const float* hrow = H   + (size_t)(row0 + l) * 16;
    for (int k0 = 0; k0 < 16; k0 += 4) {
        int kk = k0 + 2 * half;
        v2f a1; a1.x = arow[kk]; a1.y = arow[kk + 1];
        v2f b1; b1.x = cv ? Wrel[kk * OUT_F + col] : 0.0f;
                b1.y = cv ? Wrel[(kk + 1) * OUT_F + col] : 0.0f;
        c = __builtin_amdgcn_wmma_f32_16x16x4_f32(false, a1, false, b1, (short)0, c, false, false);
        v2f a2; a2.x = hrow[kk]; a2.y = hrow[kk + 1];
        v2f b2; b2.x = cv ? Wroot[kk * OUT_F + col] : 0.0f;
                b2.y = cv ? Wroot[(kk + 1) * OUT_F + col] : 0.0f;
        c = __builtin_amdgcn_wmma_f32_16x16x4_f32(false, a2, false, b2, (short)0, c, false, false);
    }
    if (cv) {
        int mb = half * 8;
        for (int i = 0; i < 8; ++i)
            OUT[(size_t)(row0 + mb + i) * OUT_F + col] = c[i];
    }
}

// ---------------------------------------------------------------------------
// Zero a float buffer (agg buffers must be re-zeroed every launch).
// ---------------------------------------------------------------------------
__global__ void zero_f32(float* __restrict__ p, int n) {
    int i = blockIdx.x * blockDim.x + threadIdx.x;
    if (i < n) p[i] = 0.0f;
}

// ---------------------------------------------------------------------------
// Edge scatter-add: agg[dst[e]][0..15] += Y[src[e]][0..15].
// One thread per edge: one 64B gather row + 16 L2 float atomics.
// edge_index layout: [2, N_EDGES] -> idx[e] = src, idx[N_EDGES + e] = dst.
// ---------------------------------------------------------------------------
__global__ void scatter16(const int* __restrict__ idx,
                          const float* __restrict__ Y,
                          float* __restrict__ agg) {
    int e = blockIdx.x * blockDim.x + threadIdx.x;
    if (e >= N_EDGES) return;
    int s = idx[e];
    int d = idx[N_EDGES + e];
    const float4* yr = (const float4*)(Y + (size_t)s * 16);
    float* ar = agg + (size_t)d * 16;
#pragma unroll
    for (int j = 0; j < 4; ++j) {
        float4 v = yr[j];
        atomicAdd(ar + 4 * j + 0, v.x);
        atomicAdd(ar + 4 * j + 1, v.y);
        atomicAdd(ar + 4 * j + 2, v.z);
        atomicAdd(ar + 4 * j + 3, v.w);
    }
}

// ---------------------------------------------------------------------------
// Elementwise combine: h = tanh(agg + root + bias[f])   over [N_NODES, 16].
// ---------------------------------------------------------------------------
__global__ void combine_tanh16(const float* __restrict__ agg,
                               const float* __restrict__ R,
                               const float* __restrict__ b,
                               float* __restrict__ h) {
    int i = blockIdx.x * blockDim.x + threadIdx.x;
    if (i < N_NODES * 16) {
        float v = agg[i] + R[i] + b[i & 15];
        h[i] = tanhf(v);
    }
}

static inline int cdiv(int a, int b) { return (a + b - 1) / b; }

extern "C" void kernel_launch(void* const* d_in, const int* in_sizes, int n_in,
                              void* d_out, int out_size, void* d_ws, size_t ws_size,
                              hipStream_t stream) {
    (void)in_sizes; (void)n_in; (void)out_size; (void)ws_size;
    const float* x       = (const float*)d_in[0];
    const int*   eidx    = (const int*)  d_in[1];   // [2, N_EDGES]
    const float* W1_rel  = (const float*)d_in[2];
    const float* b1      = (const float*)d_in[3];
    const float* W1_root = (const float*)d_in[4];
    const float* W1b_rel = (const float*)d_in[5];
    const float* b1b     = (const float*)d_in[6];
    const float* W1b_root= (const float*)d_in[7];
    const float* W2_rel  = (const float*)d_in[8];
    const float* b2      = (const float*)d_in[9];
    const float* W2_root = (const float*)d_in[10];
    float* out = (float*)d_out;

    // Workspace layout (floats): Y | R | agg | h1 | h2   (5 x 6.4 MB = 32 MB)
    const size_t NF = (size_t)N_NODES * 16;
    float* Y   = (float*)d_ws;
    float* R   = Y + NF;
    float* agg = R + NF;
    float* h1  = agg + NF;
    float* h2  = h1 + NF;

    const int gemm_threads = (N_NODES / 16) * 32;          // 6250 waves
    const int gemm_blocks  = cdiv(gemm_threads, 256);
    const int zero_blocks  = cdiv((int)NF, 256);
    const int edge_blocks  = cdiv(N_EDGES, 256);
    const int comb_blocks  = cdiv((int)NF, 256);
    const int out_blocks   = cdiv((N_NODES / 16) * 4 * 32, 256); // 25000 waves

    // ---- Layer 1: transform first (48 -> 16), then aggregate 16/edge ----
    gemm_dual16<<<gemm_blocks, 256, 0, stream>>>(x, IN_F, W1_rel, W1_root, Y, R);
    zero_f32<<<zero_blocks, 256, 0, stream>>>(agg, (int)NF);
    scatter16<<<edge_blocks, 256, 0, stream>>>(eidx, Y, agg);
    combine_tanh16<<<comb_blocks, 256, 0, stream>>>(agg, R, b1, h1);

    // ---- Layer 2 (16 -> 16) ----
    gemm_dual16<<<gemm_blocks, 256, 0, stream>>>(h1, HID_F, W1b_rel, W1b_root, Y, R);
    zero_f32<<<zero_blocks, 256, 0, stream>>>(agg, (int)NF);
    scatter16<<<edge_blocks, 256, 0, stream>>>(eidx, Y, agg);
    combine_tanh16<<<comb_blocks, 256, 0, stream>>>(agg, R, b1b, h2);

    // ---- Layer 3: aggregate first (16/edge), then fused GEMM to 60 ----
    zero_f32<<<zero_blocks, 256, 0, stream>>>(agg, (int)NF);
    scatter16<<<edge_blocks, 256, 0, stream>>>(eidx, h2, agg);
    gemm_out60<<<out_blocks, 256, 0, stream>>>(agg, h2, W2_rel, W2_root, b2, out);
}